// BilateralFilterDenoiser_61933428417260
// MI455X (gfx1250) — compile-verified
//
#include <hip/hip_runtime.h>
#include <cstdint>

// Problem constants (match reference)
#define KSZ     5
#define PAD     2
#define IMG_H   512
#define IMG_W   512

// Tiling
#define TILE_W  64
#define TILE_H  16
#define LDS_W   (TILE_W + 2 * PAD)   // 68
#define LDS_H   (TILE_H + 2 * PAD)   // 20
#define LDS_N   (LDS_W * LDS_H)      // 1360 floats
#define NTHREADS 256
#define PIX_PER_THREAD ((TILE_W * TILE_H) / NTHREADS)  // 4
#define LOAD_ITERS ((LDS_N + NTHREADS - 1) / NTHREADS) // 6

// exp(-d^2 / (2*0.1^2)) == exp2(d^2 * (-50 * log2(e)))
#define RANGE_COEF (-72.13475204444817f)

__global__ __launch_bounds__(NTHREADS)
void bilateral_kernel(const float* __restrict__ x,
                      const float* __restrict__ spatial,
                      float* __restrict__ out) {
    __shared__ float tile[LDS_N];

    const int tid = threadIdx.x;
    const int tiles_x = IMG_W / TILE_W;              // 8
    const int tiles_y = IMG_H / TILE_H;              // 32
    const int tiles_per_plane = tiles_x * tiles_y;   // 256

    const int plane = blockIdx.x / tiles_per_plane;
    const int t     = blockIdx.x % tiles_per_plane;
    const int ty0   = (t / tiles_x) * TILE_H;
    const int tx0   = (t % tiles_x) * TILE_W;

    const float* __restrict__ src = x + (size_t)plane * (IMG_H * IMG_W);
    float* __restrict__ dst       = out + (size_t)plane * (IMG_H * IMG_W);

    // ---- Async-copy halo tile (with reflect padding) into LDS ----
    // gfx1250 path: global_load_async_to_lds_b32, tracked by ASYNCcnt.
    const uint32_t lds_base = (uint32_t)(uintptr_t)(&tile[0]);   // low 32 bits = LDS byte offset
#pragma unroll
    for (int it = 0; it < LOAD_ITERS; ++it) {
        const int idx = tid + it * NTHREADS;
        if (idx < LDS_N) {
            int ly = idx / LDS_W;
            int lx = idx - ly * LDS_W;
            int gy = ty0 + ly - PAD;
            int gx = tx0 + lx - PAD;
            // jnp 'reflect' (no edge repeat): -1 -> 1, H -> H-2
            gy = (gy < 0) ? -gy : gy;
            gy = (gy >= IMG_H) ? (2 * IMG_H - 2 - gy) : gy;
            gx = (gx < 0) ? -gx : gx;
            gx = (gx >= IMG_W) ? (2 * IMG_W - 2 - gx) : gx;

            uint64_t gaddr = (uint64_t)(uintptr_t)(src + (gy * IMG_W + gx));
            uint32_t laddr = lds_base + (uint32_t)idx * 4u;
            asm volatile("global_load_async_to_lds_b32 %0, %1, off"
                         :
                         : "v"(laddr), "v"(gaddr)
                         : "memory");
        }
    }
    // Wait for this wave's async transfers, then sync the workgroup so every
    // wave sees the full tile.
    asm volatile("s_wait_asynccnt 0" ::: "memory");
    __syncthreads();

    // ---- Spatial weights: uniform, constant indices -> scalar loads ----
    float sp[KSZ * KSZ];
#pragma unroll
    for (int i = 0; i < KSZ * KSZ; ++i) sp[i] = spatial[i];

    // ---- Compute: 4 pixels per thread, lane-contiguous stores ----
#pragma unroll
    for (int i = 0; i < PIX_PER_THREAD; ++i) {
        const int p  = tid + i * NTHREADS;
        const int yy = p / TILE_W;
        const int xx = p - yy * TILE_W;

        const float c = tile[(yy + PAD) * LDS_W + (xx + PAD)];

        // Center tap: d == 0 -> range weight is exactly 1.0; no exp needed.
        float wsum = sp[(KSZ / 2) * KSZ + (KSZ / 2)];
        float psum = wsum * c;
#pragma unroll
        for (int dy = 0; dy < KSZ; ++dy) {
#pragma unroll
            for (int dx = 0; dx < KSZ; ++dx) {
                if (dy == KSZ / 2 && dx == KSZ / 2) continue;  // handled above
                const float v = tile[(yy + dy) * LDS_W + (xx + dx)];
                const float d = v - c;
                // single v_exp_f32 per tap (exp2 with folded coefficient)
                const float w = __builtin_amdgcn_exp2f(d * d * RANGE_COEF) * sp[dy * KSZ + dx];
                wsum += w;
                psum += w * v;
            }
        }
        wsum = fmaxf(wsum, 1e-10f);
        const float r = psum * __builtin_amdgcn_rcpf(wsum);  // v_rcp_f32 + v_mul
        // Output is write-once, never re-read by this dispatch: keep it out of
        // the caches that the input tiles want (th:TH_STORE_NT).
        __builtin_nontemporal_store(r, &dst[(size_t)(ty0 + yy) * IMG_W + (tx0 + xx)]);
    }
}

extern "C" void kernel_launch(void* const* d_in, const int* in_sizes, int n_in,
                              void* d_out, int out_size, void* d_ws, size_t ws_size,
                              hipStream_t stream) {
    const float* x       = (const float*)d_in[0];   // (B,C,H,W) fp32
    const float* spatial = (const float*)d_in[1];   // (5,5) fp32
    float* out           = (float*)d_out;

    const int planes = out_size / (IMG_H * IMG_W);                    // B*C = 24
    const int tiles_per_plane = (IMG_H / TILE_H) * (IMG_W / TILE_W);  // 256

    dim3 grid(planes * tiles_per_plane);
    dim3 block(NTHREADS);
    bilateral_kernel<<<grid, block, 0, stream>>>(x, spatial, out);
}